// DALayer_9783935500536
// MI455X (gfx1250) — compile-verified
//
#include <hip/hip_runtime.h>
#include <stdint.h>

typedef uint32_t v4u __attribute__((ext_vector_type(4)));
typedef int32_t  v4i __attribute__((ext_vector_type(4)));
typedef int32_t  v8i __attribute__((ext_vector_type(8)));

#define IMG_H 128
#define IMG_W 128
#define IMG_PX (IMG_H * IMG_W)   // 16384
#define NB 64
#define NO 32

__global__ __launch_bounds__(256)
void st_bilinear_kernel(const float* __restrict__ X,
                        const float* __restrict__ eps,
                        const float* __restrict__ tmin,
                        const float* __restrict__ tmax,
                        float* __restrict__ out)
{
    __shared__ float tile[IMG_PX];   // 64 KB: one full image in LDS

    const int b = blockIdx.x;        // image index   [0,64)
    const int o = blockIdx.y;        // transform idx [0,32)

    // ---- Stage image b into LDS via the Tensor Data Mover (one DMA, wave 0) ----
#if __has_builtin(__builtin_amdgcn_tensor_load_to_lds)
    if (threadIdx.x < 32) {
        const uint64_t gaddr = (uint64_t)(uintptr_t)(X + (size_t)b * IMG_PX);
        const uint32_t laddr = (uint32_t)(uintptr_t)(&tile[0]); // low 32 bits of flat LDS addr = LDS offset

        v4u g0;
        g0[0] = 1u;                                   // count=1, is_restore=0, no gather
        g0[1] = laddr;                                // lds_addr
        g0[2] = (uint32_t)(gaddr & 0xFFFFFFFFu);      // global_addr[31:0]
        g0[3] = (uint32_t)((gaddr >> 32) & 0x1FFFFFFu)// global_addr[56:32]
              | (2u << 30);                           // type=2 ("image")

        v8i g1;
        g1[0] = (int)(2u << 16);        // workgroup_mask=0, data_size=2 (4 bytes)
        g1[1] = (int)(128u << 16);      // tensor_dim0[15:0]=128 -> dword1[31:16]
        g1[2] = (int)(128u << 16);      // tensor_dim0 hi16=0 | tensor_dim1[15:0]=128
        g1[3] = (int)(128u << 16);      // tensor_dim1 hi16=0 | tile_dim0=128
        g1[4] = (int)(128u);            // tile_dim1=128, tile_dim2=0 (2-D tile)
        g1[5] = (int)(128u);            // tensor_dim0_stride = 128 elements
        g1[6] = 0;                      // stride0 hi16 = 0 | tensor_dim1_stride = 0 (unused)
        g1[7] = 0;

        v4i g2 = {0, 0, 0, 0};          // tensor_dim2/3 unused (2-D tile)
        v4i g3 = {0, 0, 0, 0};
        v8i g4 = {0, 0, 0, 0, 0, 0, 0, 0}; // trailing group: NULL (clang-23 6-arg form)

        __builtin_amdgcn_tensor_load_to_lds(g0, g1, g2, g3, g4, 0);
        __builtin_amdgcn_s_wait_tensorcnt(0);
    }
#else
    {
        const float* Xb = X + (size_t)b * IMG_PX;
        for (int i = threadIdx.x; i < IMG_PX; i += 256) tile[i] = Xb[i];
    }
#endif
    __syncthreads();

    // ---- Per-block (uniform) affine parameters for transform o ----
    float th[7];
#pragma unroll
    for (int i = 0; i < 7; ++i) {
        const float mn = tmin[i];
        const float mx = tmax[i];
        th[i] = mn + (mx - mn) * eps[o * 7 + i];
    }
    const float c = cosf(th[0]);
    const float s = sinf(th[0]);
    const float a00 = c * th[1] - s * th[4];   // c*sx - s*py
    const float a01 = c * th[3] - s * th[2];   // c*px - s*sy
    const float a10 = s * th[1] + c * th[4];   // s*sx + c*py
    const float a11 = s * th[3] + c * th[2];   // s*px + c*sy
    const float tx  = th[5];
    const float ty  = th[6];

    float* __restrict__ outp = out + (size_t)(o * NB + b) * IMG_PX;

    const float step = 2.0f / 127.0f;          // linspace(-1,1,128) spacing

    // 256 threads x 16 iters x float4 = 16384 px, fully coalesced b128 stores
#pragma unroll 4
    for (int it = 0; it < 16; ++it) {
        const int p  = it * 1024 + (int)threadIdx.x * 4;
        const int h  = p >> 7;
        const int w0 = p & 127;
        const float yg = fmaf((float)h, step, -1.0f);

        float4 res;
        float* r = &res.x;
#pragma unroll
        for (int j = 0; j < 4; ++j) {
            const int w = w0 + j;
            const float xg = fmaf((float)w, step, -1.0f);
            const float xs = fmaf(a00, xg, fmaf(a01, yg, tx));
            const float ys = fmaf(a10, xg, fmaf(a11, yg, ty));
            const float xp = (xs + 1.0f) * 63.5f;   // (W-1)*0.5
            const float yp = (ys + 1.0f) * 63.5f;

            const float x0f = floorf(xp);
            const float y0f = floorf(yp);
            const float wx = xp - x0f;
            const float wy = yp - y0f;

            int ix0 = (int)x0f;
            int iy0 = (int)y0f;
            int ix1 = ix0 + 1;
            int iy1 = iy0 + 1;
            ix0 = min(max(ix0, 0), 127);
            ix1 = min(max(ix1, 0), 127);
            iy0 = min(max(iy0, 0), 127);
            iy1 = min(max(iy1, 0), 127);

            const float Ia = tile[iy0 * IMG_W + ix0];
            const float Ib = tile[iy1 * IMG_W + ix0];
            const float Ic = tile[iy0 * IMG_W + ix1];
            const float Id = tile[iy1 * IMG_W + ix1];

            const float omx = 1.0f - wx;
            const float omy = 1.0f - wy;
            r[j] = Ia * omx * omy + Ib * omx * wy + Ic * wx * omy + Id * wx * wy;
        }
        *(float4*)(outp + p) = res;
    }
}

extern "C" void kernel_launch(void* const* d_in, const int* in_sizes, int n_in,
                              void* d_out, int out_size, void* d_ws, size_t ws_size,
                              hipStream_t stream)
{
    const float* X    = (const float*)d_in[0];   // (64, 16384)
    const float* eps  = (const float*)d_in[1];   // (32, 7)
    const float* tmin = (const float*)d_in[2];   // (1, 7)
    const float* tmax = (const float*)d_in[3];   // (1, 7)
    float* out = (float*)d_out;                  // (32*64, 128, 128, 1)

    (void)in_sizes; (void)n_in; (void)out_size; (void)d_ws; (void)ws_size;

    dim3 grid(NB, NO);   // one workgroup per (image, transform)
    dim3 block(256);
    st_bilinear_kernel<<<grid, block, 0, stream>>>(X, eps, tmin, tmax, out);
}